// WTAInhibition_56049323213388
// MI455X (gfx1250) — compile-verified
//
#include <hip/hip_runtime.h>
#include <cstdint>

// Shapes fixed by the reference: B=16, C=64, H=W=128.
#define B_  16
#define C_  64
#define H_  128
#define W_  128
#define HW_ 16384           // H_*W_
#define N_  16777216        // B_*C_*HW_
#define CHUNK 2048          // floats per array per pipeline stage
#define NIT (HW_ / CHUNK)   // 8 stages per plane

// ---------------------------------------------------------------------------
// Pass 1: per-(b,c) masked argmax, streamed via CDNA5 async global->LDS DMA.
// Each lane async-loads its own 8-float slice of each array into LDS and
// consumes exactly that slice, so no barriers are needed inside the pipeline;
// ASYNCcnt ordering (loads complete in order) makes s_wait_asynccnt 4 wait
// precisely for the current chunk while the next chunk is in flight.
// ---------------------------------------------------------------------------
__global__ void __launch_bounds__(256)
wta_argmax_kernel(const float* __restrict__ spikes,
                  const float* __restrict__ prm,
                  int* __restrict__ widx)
{
    __shared__ __align__(16) float s_sp[2][CHUNK];
    __shared__ __align__(16) float s_pm[2][CHUNK];
    __shared__ float r_v[8];
    __shared__ int   r_i[8];

    const int bc = blockIdx.x;           // 0..1023  (b*64 + c)
    const int t  = threadIdx.x;          // 0..255
    const float* gsp = spikes + (size_t)bc * HW_ + t * 8;
    const float* gpm = prm    + (size_t)bc * HW_ + t * 8;

    auto issue = [&](int it, int buf) {
        unsigned l0 = (unsigned)(uintptr_t)&s_sp[buf][t * 8];
        unsigned l1 = (unsigned)(uintptr_t)&s_pm[buf][t * 8];
        const float* g0 = gsp + it * CHUNK;
        const float* g1 = gpm + it * CHUNK;
        // INST_OFFSET applies to both the global and LDS address (ISA 15.18.3).
        asm volatile("global_load_async_to_lds_b128 %0, %1, off"
                     :: "v"(l0), "v"(g0) : "memory");
        asm volatile("global_load_async_to_lds_b128 %0, %1, off offset:16"
                     :: "v"(l0), "v"(g0) : "memory");
        asm volatile("global_load_async_to_lds_b128 %0, %1, off"
                     :: "v"(l1), "v"(g1) : "memory");
        asm volatile("global_load_async_to_lds_b128 %0, %1, off offset:16"
                     :: "v"(l1), "v"(g1) : "memory");
    };

    float bv = -INFINITY;
    int   bi = 0x7FFFFFFF;

    issue(0, 0);
    for (int it = 0; it < NIT; ++it) {
        const int buf = it & 1;
        if (it + 1 < NIT) {
            issue(it + 1, buf ^ 1);                        // prefetch next stage
            asm volatile("s_wait_asynccnt 4" ::: "memory"); // current stage landed
        } else {
            asm volatile("s_wait_asynccnt 0" ::: "memory");
        }
        const float4 a0 = *reinterpret_cast<const float4*>(&s_sp[buf][t * 8]);
        const float4 a1 = *reinterpret_cast<const float4*>(&s_sp[buf][t * 8 + 4]);
        const float4 p0 = *reinterpret_cast<const float4*>(&s_pm[buf][t * 8]);
        const float4 p1 = *reinterpret_cast<const float4*>(&s_pm[buf][t * 8 + 4]);
        const float sarr[8] = {a0.x, a0.y, a0.z, a0.w, a1.x, a1.y, a1.z, a1.w};
        const float parr[8] = {p0.x, p0.y, p0.z, p0.w, p1.x, p1.y, p1.z, p1.w};
        const int base = it * CHUNK + t * 8;
#pragma unroll
        for (int j = 0; j < 8; ++j) {
            const float v = (sarr[j] > 0.0f) ? parr[j] : -INFINITY;
            const int idx = base + j;
            if (v > bv || (v == bv && idx < bi)) { bv = v; bi = idx; }
        }
    }

    // wave32 shuffle reduction (first-index tie-break == jnp.argmax)
#pragma unroll
    for (int off = 16; off > 0; off >>= 1) {
        const float ov = __shfl_down(bv, off, 32);
        const int   oi = __shfl_down(bi, off, 32);
        if (ov > bv || (ov == bv && oi < bi)) { bv = ov; bi = oi; }
    }
    const int wave = t >> 5, lane = t & 31;
    if (lane == 0) { r_v[wave] = bv; r_i[wave] = bi; }
    __syncthreads();
    if (t == 0) {
        for (int wv = 1; wv < 8; ++wv) {
            const float ov = r_v[wv]; const int oi = r_i[wv];
            if (ov > bv || (ov == bv && oi < bi)) { bv = ov; bi = oi; }
        }
        widx[bc] = (bv == -INFINITY) ? -1 : bi;   // -1 <=> channel never fired
    }
}

// ---------------------------------------------------------------------------
// Pass 2: per-(b,h,w) dilated any-winner map. 64 winner indices per batch are
// wave-uniform and L2-hot; Chebyshev-distance test replaces reduce_window.
// ---------------------------------------------------------------------------
__global__ void __launch_bounds__(256)
wta_zone_kernel(const int* __restrict__ widx,
                const int* __restrict__ rad,
                int* __restrict__ zone)
{
    const int t  = blockIdx.x * blockDim.x + threadIdx.x;  // 0..B_*HW_-1
    const int b  = t >> 14;
    const int hw = t & (HW_ - 1);
    const int h  = hw >> 7, w = hw & (W_ - 1);
    const int r  = rad[0];
    const int* wb = widx + b * C_;
    int z = 0;
#pragma unroll 8
    for (int c = 0; c < C_; ++c) {
        const int wi = wb[c];
        if (wi >= 0) {
            const int dh = (wi >> 7) - h;
            const int dw = (wi & (W_ - 1)) - w;
            z |= (dh <= r && dh >= -r && dw <= r && dw >= -r) ? 1 : 0;
        }
    }
    zone[t] = z;
}

// ---------------------------------------------------------------------------
// Pass 3: emit all three outputs, float4 per thread (b128 loads/stores).
// new_spikes == winner_mask; membrane zeroed if channel fired OR in zone.
// ---------------------------------------------------------------------------
__global__ void __launch_bounds__(256)
wta_finalize_kernel(const float* __restrict__ membrane,
                    const int* __restrict__ widx,
                    const int* __restrict__ zone,
                    float* __restrict__ out)
{
    const int t    = blockIdx.x * blockDim.x + threadIdx.x;  // 0..N_/4-1
    const int base = t * 4;
    const int bc   = base >> 14;
    const int hw   = base & (HW_ - 1);
    const int b    = bc >> 6;

    const int wi = widx[bc];                       // wave-uniform
    const int4   zn = *reinterpret_cast<const int4*>(zone + b * HW_ + hw);
    const float4 m  = *reinterpret_cast<const float4*>(membrane + base);
    const bool reset = (wi >= 0);

    float4 sp, nm;
    sp.x = (wi == hw + 0) ? 1.0f : 0.0f;
    sp.y = (wi == hw + 1) ? 1.0f : 0.0f;
    sp.z = (wi == hw + 2) ? 1.0f : 0.0f;
    sp.w = (wi == hw + 3) ? 1.0f : 0.0f;
    nm.x = (reset || zn.x) ? 0.0f : m.x;
    nm.y = (reset || zn.y) ? 0.0f : m.y;
    nm.z = (reset || zn.z) ? 0.0f : m.z;
    nm.w = (reset || zn.w) ? 0.0f : m.w;

    *reinterpret_cast<float4*>(out + base)            = sp;  // new_spikes
    *reinterpret_cast<float4*>(out + N_ + base)       = nm;  // new_membrane
    *reinterpret_cast<float4*>(out + 2 * N_ + base)   = sp;  // winner_mask
}

// ---------------------------------------------------------------------------
extern "C" void kernel_launch(void* const* d_in, const int* in_sizes, int n_in,
                              void* d_out, int out_size, void* d_ws, size_t ws_size,
                              hipStream_t stream)
{
    const float* spikes   = (const float*)d_in[0];
    const float* membrane = (const float*)d_in[1];
    const float* prm      = (const float*)d_in[2];
    const int*   rad      = (const int*)d_in[3];   // scalar radius (device)
    float* out = (float*)d_out;

    int* widx = (int*)d_ws;          // 1024 ints (fully written each call)
    int* zone = widx + (B_ * C_);    // 262144 ints (fully written each call)

    wta_argmax_kernel  <<<B_ * C_,            256, 0, stream>>>(spikes, prm, widx);
    wta_zone_kernel    <<<(B_ * HW_) / 256,   256, 0, stream>>>(widx, rad, zone);
    wta_finalize_kernel<<<(N_ / 4) / 256,     256, 0, stream>>>(membrane, widx, zone, out);
}